// GAT_6390911337028
// MI455X (gfx1250) — compile-verified
//
#include <hip/hip_runtime.h>
#include <hip/hip_bf16.h>

typedef __attribute__((ext_vector_type(2))) float v2f;
typedef __attribute__((ext_vector_type(8))) float v8f;

#define NNODES 50000
#define NEDGES 800000
#define NEG_SLOPE 0.2f

// ---------------- atomic float max (segment_max) ----------------
__device__ __forceinline__ void atomicMaxF(float* addr, float val) {
  // int-max for non-negative, uint-min for negative: totally ordered on IEEE bits
  if (val >= 0.0f) atomicMax((int*)addr, __float_as_int(val));
  else             atomicMin((unsigned int*)addr, __float_as_uint(val));
}

// ---------------- generic fill ----------------
__global__ void k_fill(float* __restrict__ p, float v, long n) {
  long i = (long)blockIdx.x * blockDim.x + threadIdx.x;
  if (i < n) p[i] = v;
}

// ---------------- B repack: B[K,NCOL] -> Bp[K/2][NCOL][2] ----------------
// Makes each WMMA B-fragment K-pair contiguous so it loads as one b64.
__global__ void k_repackB(const float* __restrict__ B, float* __restrict__ Bp,
                          int K, int NCOL) {
  int i = blockIdx.x * blockDim.x + threadIdx.x;
  if (i >= K * NCOL) return;
  int k = i / NCOL, n = i % NCOL;
  Bp[((long)(k >> 1) * NCOL + n) * 2 + (k & 1)] = B[i];
}

// ---------------- fp32 WMMA GEMM: C[M,NCOL] = A[M,K] @ B[K,NCOL] ----------------
// One wave computes a 64-row x 64-col tile: 4 A fragments x 4 B fragments =
// 16 v_wmma_f32_16x16x4_f32 per K-step. K/NCOL are template constants so all
// fragment addresses are one base pointer + immediate offsets (no spills).
// A fragment (16x4 f32): lane L holds row M=L&15, K pair koff = 2*(L>>4).
// C/D (16x16 f32, 8 VGPRs): vgpr r -> row r (lanes 0-15) / row r+8 (lanes 16-31).
template<int K, int NCOL>
__global__ __launch_bounds__(32, 1)
void k_wmma_gemm(const float* __restrict__ A, const float* __restrict__ Bp,
                 float* __restrict__ C, int nRowTiles) {
  const int lane = threadIdx.x & 31;
  int rt0 = blockIdx.y << 2;                 // 4 row tiles (64 rows) per wave
  if (rt0 + 4 > nRowTiles) rt0 = nRowTiles - 4;   // shift last block down:
  // overlapped tiles recompute identical values -> duplicate stores are benign,
  // and EXEC stays all-1s with zero store guards.
  const int cg0  = blockIdx.x << 6;          // 64 output columns per wave
  const int koff = (lane >> 4) << 1;         // 0 or 2
  const int nb   = cg0 + (lane & 15);

  const float* Abase = A + (long)(rt0 * 16 + (lane & 15)) * K + koff;
  const float* Bbase = Bp + (long)nb * 2 + koff * NCOL;   // koff/2 pairs * NCOL * 2

  v8f acc[4][4];
  #pragma unroll
  for (int f = 0; f < 4; f++)
    #pragma unroll
    for (int j = 0; j < 4; j++) acc[f][j] = (v8f){};

  #pragma unroll 4
  for (int k = 0; k < K; k += 4) {
    v2f a[4];
    #pragma unroll
    for (int f = 0; f < 4; f++)
      a[f] = *(const v2f*)(Abase + k + f * 16 * K);        // b64, imm offsets
    const float* Bk = Bbase + (long)(k >> 1) * NCOL * 2;   // pair row
    v2f b[4];
    #pragma unroll
    for (int j = 0; j < 4; j++)
      b[j] = *(const v2f*)(Bk + j * 32);                   // b64, imm offsets
    #pragma unroll
    for (int f = 0; f < 4; f++)
      #pragma unroll
      for (int j = 0; j < 4; j++)
        acc[f][j] = __builtin_amdgcn_wmma_f32_16x16x4_f32(
            false, a[f], false, b[j], (short)0, acc[f][j], false, false);
  }

  const int colW = cg0 + (lane & 15);
  #pragma unroll
  for (int f = 0; f < 4; f++) {
    int rowW = (rt0 + f) * 16 + ((lane >> 4) << 3);
    #pragma unroll
    for (int r = 0; r < 8; r++) {
      float* Cp = C + (long)(rowW + r) * NCOL + colW;
      Cp[0]  = acc[f][0][r];
      Cp[16] = acc[f][1][r];
      Cp[32] = acc[f][2][r];
      Cp[48] = acc[f][3][r];
    }
  }
}

// ---------------- per-(node,head) attention scores ----------------
__global__ void k_attn(const float* __restrict__ h, const float* __restrict__ a_src,
                       const float* __restrict__ a_dst, float* __restrict__ asrc,
                       float* __restrict__ adst, int N, int H, int C) {
  int i = blockIdx.x * blockDim.x + threadIdx.x;   // n*H + head
  if (i >= N * H) return;
  int n = i / H, hh = i % H;
  const float* hp = h + (long)n * H * C + (long)hh * C;
  const float* as = a_src + hh * C;
  const float* ad = a_dst + hh * C;
  float s1 = 0.f, s2 = 0.f;
  for (int c = 0; c < C; c++) { float v = hp[c]; s1 += v * as[c]; s2 += v * ad[c]; }
  asrc[i] = s1; adst[i] = s2;
}

__device__ __forceinline__ float leaky(float x) { return x > 0.f ? x : NEG_SLOPE * x; }

// ---------------- edge pass 1: segment max of leaky logits ----------------
__global__ void k_edge_max(const int* __restrict__ src, const int* __restrict__ dst,
                           const float* __restrict__ asrc, const float* __restrict__ adst,
                           float* __restrict__ m, int H) {
  long i = (long)blockIdx.x * blockDim.x + threadIdx.x;
  if (i >= (long)NEDGES * H) return;
  int e = (int)(i / H), hh = (int)(i % H);
  int s = src[e], d = dst[e];
  atomicMaxF(&m[d * H + hh], leaky(asrc[s * H + hh] + adst[d * H + hh]));
}

// ---------------- edge pass 2: segment sum of exp(logit - m[dst]) ----------------
__global__ void k_edge_den(const int* __restrict__ src, const int* __restrict__ dst,
                           const float* __restrict__ asrc, const float* __restrict__ adst,
                           const float* __restrict__ m, float* __restrict__ den, int H) {
  long i = (long)blockIdx.x * blockDim.x + threadIdx.x;
  if (i >= (long)NEDGES * H) return;
  int e = (int)(i / H), hh = (int)(i % H);
  int s = src[e], d = dst[e];
  float lg = leaky(asrc[s * H + hh] + adst[d * H + hh]);
  atomicAdd(&den[d * H + hh], expf(lg - m[d * H + hh]));
}

// ---------------- edge pass 3: weighted scatter-aggregate ----------------
// thread = (edge, group of 4 channels); h fits in L2 so gather + atomics stay on-chip
__global__ void k_edge_agg(const int* __restrict__ src, const int* __restrict__ dst,
                           const float* __restrict__ asrc, const float* __restrict__ adst,
                           const float* __restrict__ m, const float* __restrict__ den,
                           const float* __restrict__ hfeat, float* __restrict__ agg,
                           int H, int C) {
  const int HC = H * C;
  long i = (long)blockIdx.x * blockDim.x + threadIdx.x;
  long total = (long)NEDGES * (HC >> 2);
  if (i >= total) return;
  int e  = (int)(i / (HC >> 2));
  int c0 = ((int)(i % (HC >> 2))) << 2;
  int hh = c0 / C;
  int s = src[e], d = dst[e];
  float lg = leaky(asrc[s * H + hh] + adst[d * H + hh]);
  float alpha = expf(lg - m[d * H + hh]) / (den[d * H + hh] + 1e-16f);
  const float4 hv = *(const float4*)(hfeat + (long)s * HC + c0);
  float* out = agg + (long)d * HC + c0;
  atomicAdd(out + 0, alpha * hv.x);
  atomicAdd(out + 1, alpha * hv.y);
  atomicAdd(out + 2, alpha * hv.z);
  atomicAdd(out + 3, alpha * hv.w);
}

// ---------------- epilogue: bias + optional ELU (in place ok) ----------------
__global__ void k_epi(const float* __restrict__ agg, const float* __restrict__ bias,
                      float* __restrict__ out, int HC, int do_elu) {
  long i = (long)blockIdx.x * blockDim.x + threadIdx.x;
  if (i >= (long)NNODES * HC) return;
  float v = agg[i] + bias[(int)(i % HC)];
  if (do_elu) v = v > 0.f ? v : (expf(v) - 1.0f);
  out[i] = v;
}

template<int K, int HC>
static void run_layer(const float* in, const float* W, const float* avs, const float* avd,
                      const float* bias, int Hh, int Cc,
                      float* Hbuf, float* AGG, float* BP,
                      float* ASRC, float* ADST, float* Mx, float* DEN,
                      const int* src, const int* dst, float* outbuf, int do_elu,
                      hipStream_t stream) {
  const int nRowTiles = NNODES / 16;                 // 3125
  k_repackB<<<(K * HC + 255) / 256, 256, 0, stream>>>(W, BP, K, HC);
  dim3 gg(HC / 64, (nRowTiles + 3) / 4);             // 64x64 tile per wave
  k_wmma_gemm<K, HC><<<gg, dim3(32), 0, stream>>>(in, BP, Hbuf, nRowTiles);
  int nh = NNODES * Hh;
  k_attn<<<(nh + 255) / 256, 256, 0, stream>>>(Hbuf, avs, avd, ASRC, ADST, NNODES, Hh, Cc);
  long na = (long)NNODES * HC;
  k_fill<<<(int)((na + 255) / 256), 256, 0, stream>>>(AGG, 0.0f, na);  // safe: GEMM already consumed AGG-aliased input
  k_fill<<<(nh + 255) / 256, 256, 0, stream>>>(Mx, -__builtin_inff(), nh);
  k_fill<<<(nh + 255) / 256, 256, 0, stream>>>(DEN, 0.0f, nh);
  long eh = (long)NEDGES * Hh;
  k_edge_max<<<(int)((eh + 255) / 256), 256, 0, stream>>>(src, dst, ASRC, ADST, Mx, Hh);
  k_edge_den<<<(int)((eh + 255) / 256), 256, 0, stream>>>(src, dst, ASRC, ADST, Mx, DEN, Hh);
  long et = (long)NEDGES * (HC / 4);
  k_edge_agg<<<(int)((et + 255) / 256), 256, 0, stream>>>(src, dst, ASRC, ADST, Mx, DEN, Hbuf, AGG, Hh, Cc);
  k_epi<<<(int)((na + 255) / 256), 256, 0, stream>>>(AGG, bias, outbuf, HC, do_elu);
}

extern "C" void kernel_launch(void* const* d_in, const int* in_sizes, int n_in,
                              void* d_out, int out_size, void* d_ws, size_t ws_size,
                              hipStream_t stream) {
  const float* x   = (const float*)d_in[0];
  const int*   ei  = (const int*)d_in[1];
  const int*   src = ei;              // edge_index[0]
  const int*   dst = ei + NEDGES;     // edge_index[1]
  const float* W1  = (const float*)d_in[2];
  const float* as1 = (const float*)d_in[3];
  const float* ad1 = (const float*)d_in[4];
  const float* b1  = (const float*)d_in[5];
  const float* W2  = (const float*)d_in[6];
  const float* as2 = (const float*)d_in[7];
  const float* ad2 = (const float*)d_in[8];
  const float* b2  = (const float*)d_in[9];
  const float* W3  = (const float*)d_in[10];
  const float* as3 = (const float*)d_in[11];
  const float* ad3 = (const float*)d_in[12];
  const float* b3  = (const float*)d_in[13];

  // workspace layout (floats): F doubles as AGG (GEMM input is dead before AGG zero-fill)
  float* ws   = (float*)d_ws;
  float* F    = ws;                             // N*256
  float* H    = F + (long)NNODES * 256;         // N*256
  float* ASRC = H + (long)NNODES * 256;         // N*8
  float* ADST = ASRC + NNODES * 8;              // N*8
  float* Mx   = ADST + NNODES * 8;              // N*8
  float* DEN  = Mx + NNODES * 8;                // N*8
  float* BP   = DEN + NNODES * 8;               // 256*256 (repacked weights)

  // layer 1: x[N,128] -> F[N,256] (8 heads x 32, concat, ELU)
  run_layer<128, 256>(x, W1, as1, ad1, b1, 8, 32, H, F, BP, ASRC, ADST, Mx, DEN,
                      src, dst, F, 1, stream);
  // layer 2: F[N,256] -> F[N,256] (8 heads x 32, concat, ELU)
  run_layer<256, 256>(F, W2, as2, ad2, b2, 8, 32, H, F, BP, ASRC, ADST, Mx, DEN,
                      src, dst, F, 1, stream);
  // layer 3: F[N,256] -> out[N,64] (1 head x 64, no concat, no ELU)
  run_layer<256, 64>(F, W3, as3, ad3, b3, 1, 64, H, F, BP, ASRC, ADST, Mx, DEN,
                     src, dst, (float*)d_out, 0, stream);
}